// FOF_Normal_40389872451730
// MI455X (gfx1250) — compile-verified
//
#include <hip/hip_runtime.h>
#include <stdint.h>

// ---------------------------------------------------------------------------
// FOF rasterizer for MI455X (gfx1250, wave32).
// d_out layout (floats): [res BHH | dF BHH | dB BHH | nF 3BHH | nB 3BHH]
// dF/dB sections double as order-preserving uint32 depth-encoding arrays
// during the atomic passes and are decoded in place by the finalize pass.
// ---------------------------------------------------------------------------

typedef __attribute__((ext_vector_type(4))) unsigned int u32x4;
typedef __attribute__((ext_vector_type(8))) int          i32x8;
typedef __attribute__((ext_vector_type(4))) int          i32x4;

// Monotonic bijection float -> uint32 (total order matches float <).
__device__ __forceinline__ unsigned encf(float f) {
  unsigned u = __float_as_uint(f);
  return (u & 0x80000000u) ? ~u : (u | 0x80000000u);
}
__device__ __forceinline__ float decf(unsigned e) {
  return __uint_as_float((e & 0x80000000u) ? (e & 0x7FFFFFFFu) : ~e);
}

// Async Tensor-Data-Mover prefetch of `ndw` dwords at gsrc into LDS scratch.
// 1-D tile: data_size=4B, tensor_dim0 = tile_dim0 = ndw, tensor_dim1 = 1.
// Group0: count=1 | lds_addr | global_addr[31:0] | global_addr[56:32] + type=2.
// This toolchain's builtin takes 6 args (g0, g1, g2, g3, extra, cpol).
__device__ __forceinline__ void tdm_prefetch(const void* gsrc, unsigned ndw,
                                             unsigned lds_off) {
  uint64_t ga = (uint64_t)(uintptr_t)gsrc;
  u32x4 g0 = {1u, lds_off, (unsigned)ga,
              (unsigned)((ga >> 32) & 0x1FFFFFFu) | 0x80000000u};
  i32x8 g1;
  g1[0] = (int)(2u << 16);                                  // data_size = 4B
  g1[1] = (int)((ndw & 0xFFFFu) << 16);                     // tensor_dim0 lo
  g1[2] = (int)(((ndw >> 16) & 0xFFFFu) | (1u << 16));      // dim0 hi | dim1=1
  g1[3] = (int)((ndw & 0xFFFFu) << 16);                     // tile_dim0
  g1[4] = 0;                                                // tile_dim1/2 unused
  g1[5] = (int)ndw;                                         // dim0_stride lo
  g1[6] = 0;
  g1[7] = 0;
  i32x4 gz4 = {0, 0, 0, 0};
  i32x8 gz8 = {0, 0, 0, 0, 0, 0, 0, 0};
  __builtin_amdgcn_tensor_load_to_lds(g0, g1, gz4, gz4, gz8, 0);
}

// ---- Pass 0: initialize all output sections ------------------------------
__global__ void k_init(unsigned* __restrict__ out_u, int npix) {
  int i = blockIdx.x * blockDim.x + threadIdx.x;
  int total = 9 * npix;
  if (i >= total) return;
  // dF-encoding section gets +inf sentinel (max uint), everything else 0.
  out_u[i] = (i >= npix && i < 2 * npix) ? 0xFFFFFFFFu : 0u;
}

// ---- Pass 1: depth min/max via encoded integer atomics -------------------
__global__ void k_depth(const float* __restrict__ v, const float* __restrict__ vn,
                        const int* __restrict__ C, const int* __restrict__ Hp,
                        float* out, int npix, int npts, int F) {
  const int H = *Hp;
  const int t = blockIdx.x * blockDim.x + threadIdx.x;

  // Async TDM prefetch of this block's vertex slice into LDS scratch
  // (wave 0 issues once per block; real loads below then hit LDS-warmed L2).
  __shared__ unsigned tdm_buf[768];
  if ((threadIdx.x >> 5) == 0) {
    long long blk0 = (long long)blockIdx.x * blockDim.x;
    if (blk0 < npts) {
      long long rem  = (long long)npts - blk0;
      long long take = rem < (long long)blockDim.x ? rem : (long long)blockDim.x;
      tdm_prefetch(v + blk0 * 3, (unsigned)take * 3u,
                   (unsigned)(uintptr_t)tdm_buf);
    }
  }

  if (t < npts) {
    const int f3 = 3 * F;
    const int b = t / f3;
    const int rem = t - b * f3;
    const int f = rem / 3;
    const float px = v[3 * t + 0], py = v[3 * t + 1], pz = v[3 * t + 2];
    const float Hf = (float)H, ha = 0.5f * Hf, hb = ha - 0.5f;
    const float sx = fmaf(px, ha, hb);
    const float sy = fmaf(py, -ha, hb);
    const int ix = (int)floorf(sx);
    const int iy = (int)floorf(sy);
    if ((f < C[b]) && ix >= 0 && ix < H && iy >= 0 && iy < H) {
      const int pid = b * (H * H) + iy * H + ix;
      const unsigned e = encf(pz);
      unsigned* dFe = (unsigned*)out + npix;
      unsigned* dBe = (unsigned*)out + 2 * npix;
      atomicMin(&dFe[pid], e);   // global_atomic_min_u32 (no-return)
      atomicMax(&dBe[pid], e);   // global_atomic_max_u32 (no-return)
      out[pid] = 1.0f;           // coverage flag (idempotent plain store)
    }
    // Warm the 192MB L2 with the normals this thread will need in pass 2.
    __builtin_prefetch(&vn[3 * t], 0, 1);   // global_prefetch_b8
  }
  if ((threadIdx.x >> 5) == 0) __builtin_amdgcn_s_wait_tensorcnt(0);
}

// ---- Pass 2: conditional normal accumulation -----------------------------
__global__ void k_normal(const float* __restrict__ v, const float* __restrict__ vn,
                         const int* __restrict__ C, const int* __restrict__ Hp,
                         float* out, int npix, int npts, int F) {
  const int H = *Hp;
  const int t = blockIdx.x * blockDim.x + threadIdx.x;

  __shared__ unsigned tdm_buf[768];
  if ((threadIdx.x >> 5) == 0) {
    long long blk0 = (long long)blockIdx.x * blockDim.x;
    if (blk0 < npts) {
      long long rem  = (long long)npts - blk0;
      long long take = rem < (long long)blockDim.x ? rem : (long long)blockDim.x;
      tdm_prefetch(vn + blk0 * 3, (unsigned)take * 3u,
                   (unsigned)(uintptr_t)tdm_buf);
    }
  }

  if (t < npts) {
    const int f3 = 3 * F;
    const int b = t / f3;
    const int rem = t - b * f3;
    const int f = rem / 3;
    const float px = v[3 * t + 0], py = v[3 * t + 1], pz = v[3 * t + 2];
    const float Hf = (float)H, ha = 0.5f * Hf, hb = ha - 0.5f;
    const float sx = fmaf(px, ha, hb);
    const float sy = fmaf(py, -ha, hb);
    const int ix = (int)floorf(sx);
    const int iy = (int)floorf(sy);
    if ((f < C[b]) && ix >= 0 && ix < H && iy >= 0 && iy < H) {
      const int HH = H * H;
      const int pix = iy * H + ix;
      const int pid = b * HH + pix;
      const unsigned* dFe = (const unsigned*)out + npix;
      const unsigned* dBe = (const unsigned*)out + 2 * npix;
      const float zmin = decf(dFe[pid]);
      const float zmax = decf(dBe[pid]);
      const bool isF = (pz == zmin), isB = (pz == zmax);
      if (isF || isB) {
        const float nx = vn[3 * t + 0], ny = vn[3 * t + 1], nz = vn[3 * t + 2];
        float* nF = out + (size_t)3 * npix;   // (B,3,H,H) channel-major
        float* nB = out + (size_t)6 * npix;
        const size_t base = (size_t)(b * 3) * HH + pix;
        if (isF) {
          unsafeAtomicAdd(&nF[base], nx);            // global_atomic_add_f32
          unsafeAtomicAdd(&nF[base + HH], ny);
          unsafeAtomicAdd(&nF[base + 2 * (size_t)HH], nz);
        }
        if (isB) {
          unsafeAtomicAdd(&nB[base], nx);
          unsafeAtomicAdd(&nB[base + HH], ny);
          unsafeAtomicAdd(&nB[base + 2 * (size_t)HH], nz);
        }
      }
    }
  }
  if ((threadIdx.x >> 5) == 0) __builtin_amdgcn_s_wait_tensorcnt(0);
}

// ---- Pass 3: decode depths, mask, normalize normals in place -------------
__global__ void k_final(float* out, int npix, int HH) {
  const int p = blockIdx.x * blockDim.x + threadIdx.x;
  if (p >= npix) return;
  unsigned* dFe = (unsigned*)out + npix;
  unsigned* dBe = (unsigned*)out + 2 * npix;
  const bool cov = out[p] != 0.0f;
  const unsigned eF = dFe[p], eB = dBe[p];
  ((float*)dFe)[p] = cov ? decf(eF) : 0.0f;
  ((float*)dBe)[p] = cov ? decf(eB) : 0.0f;

  const int b = p / HH, pix = p - b * HH;
  float* nF = out + (size_t)3 * npix;
  float* nB = out + (size_t)6 * npix;
  const size_t i0 = (size_t)(b * 3) * HH + pix;
  const size_t i1 = i0 + HH, i2 = i1 + HH;
  {
    float x = nF[i0], y = nF[i1], z = nF[i2];   // uncovered pixels are 0 (init)
    float tn = sqrtf(x * x + y * y + z * z);
    float d = (tn == 0.0f) ? 1.0f : tn;
    nF[i0] = x / d; nF[i1] = y / d; nF[i2] = z / d;
  }
  {
    float x = nB[i0], y = nB[i1], z = nB[i2];
    float tn = sqrtf(x * x + y * y + z * z);
    float d = (tn == 0.0f) ? 1.0f : tn;
    nB[i0] = x / d; nB[i1] = y / d; nB[i2] = z / d;
  }
}

// ---------------------------------------------------------------------------
extern "C" void kernel_launch(void* const* d_in, const int* in_sizes, int n_in,
                              void* d_out, int out_size, void* d_ws, size_t ws_size,
                              hipStream_t stream) {
  (void)n_in; (void)d_ws; (void)ws_size;
  const float* v  = (const float*)d_in[0];
  const float* vn = (const float*)d_in[1];
  const int*   C  = (const int*)d_in[2];
  const int*   Hp = (const int*)d_in[3];

  const int B    = in_sizes[2];
  const int npts = in_sizes[0] / 3;        // B * F * 3 vertices (points)
  const int F    = in_sizes[0] / (9 * B);
  const int npix = out_size / 9;           // B * H * H
  const int HH   = npix / B;               // H * H

  const int TB = 256;
  k_init  <<<(9 * npix + TB - 1) / TB, TB, 0, stream>>>((unsigned*)d_out, npix);
  k_depth <<<(npts + TB - 1) / TB, TB, 0, stream>>>(v, vn, C, Hp, (float*)d_out,
                                                    npix, npts, F);
  k_normal<<<(npts + TB - 1) / TB, TB, 0, stream>>>(v, vn, C, Hp, (float*)d_out,
                                                    npix, npts, F);
  k_final <<<(npix + TB - 1) / TB, TB, 0, stream>>>((float*)d_out, npix, HH);
}